// MarginLoss_48189533061232
// MI455X (gfx1250) — compile-verified
//
#include <hip/hip_runtime.h>
#include <math.h>

typedef __attribute__((ext_vector_type(2))) float v2f;
typedef __attribute__((ext_vector_type(8))) float v8f;

#define MARGIN_F 0.2f
#define EPS_F    1e-8f
#define DIM      128
#define TRIPS_PER_WAVE  16
#define WAVES_PER_BLOCK 8
#define BLOCK_THREADS   (WAVES_PER_BLOCK * 32)
#define TRIPS_PER_BLOCK (WAVES_PER_BLOCK * TRIPS_PER_WAVE)

// Select acc[j] for dynamic j in [0,8) with a cndmask chain (stays in VGPRs).
__device__ __forceinline__ float diag_pick(const v8f& acc, int j) {
    float v = acc[0];
#pragma unroll
    for (int r = 1; r < 8; ++r) v = (j == r) ? acc[r] : v;
    return v;
}

__device__ __forceinline__ float wave_reduce_add(float v) {
#pragma unroll
    for (int off = 16; off > 0; off >>= 1)
        v += __shfl_xor(v, off, 32);
    return v;
}

__global__ __launch_bounds__(BLOCK_THREADS)
void margin_loss_partial(const float* __restrict__ batch,
                         const int*   __restrict__ triplets,
                         const int*   __restrict__ labels_idx,
                         const float* __restrict__ beta_param,
                         float*       __restrict__ partials /* [2*gridDim.x] */)
{
    const int tid  = threadIdx.x;
    const int lane = tid & 31;
    const int wave = tid >> 5;
    const int t0   = (blockIdx.x * WAVES_PER_BLOCK + wave) * TRIPS_PER_WAVE;

    // WMMA f32 16x16x4 A-operand layout: lanes 0-15 feed row m=lane with
    // K-pair {k0,k0+1}; lanes 16-31 feed row m=lane-16 with K-pair {k0+2,k0+3}.
    // B-operand (4x16) uses the identical lane<->(col,K-pair) mapping, so the
    // same registers serve as both A and B => Gram matrix; diagonal = ||d||^2.
    const int mrow    = lane & 15;          // tile row == triplet t0 + mrow
    const int ti_load = t0 + mrow;
    const int ia  = triplets[3 * ti_load + 0];
    const int ip  = triplets[3 * ti_load + 1];
    const int in_ = triplets[3 * ti_load + 2];
    const int kin = (lane >> 4) * 2;        // K-pair offset within chunk: 0 or 2

    const float* arow = batch + (size_t)ia  * DIM + kin;
    const float* prow = batch + (size_t)ip  * DIM + kin;
    const float* nrow = batch + (size_t)in_ * DIM + kin;

    v8f accAP = {0.f,0.f,0.f,0.f,0.f,0.f,0.f,0.f};
    v8f accAN = {0.f,0.f,0.f,0.f,0.f,0.f,0.f,0.f};

#pragma unroll 4
    for (int k0 = 0; k0 < DIM; k0 += 4) {
        v2f a2 = *(const v2f*)(arow + k0);
        v2f p2 = *(const v2f*)(prow + k0);
        v2f n2 = *(const v2f*)(nrow + k0);
        v2f dap = a2 - p2;
        v2f dan = a2 - n2;
        // D = dap * dap^T (+C): diagonal accumulates ||a_i - p_i||^2 in f32.
        accAP = __builtin_amdgcn_wmma_f32_16x16x4_f32(
            false, dap, false, dap, (short)0, accAP, false, false);
        accAN = __builtin_amdgcn_wmma_f32_16x16x4_f32(
            false, dan, false, dan, (short)0, accAN, false, false);
    }

    // C/D layout: VGPR r holds M=r (lanes 0-15, N=lane) and M=r+8
    // (lanes 16-31, N=lane-16). Diagonal (i,i): i<8 -> lane i, VGPR i;
    // i>=8 -> lane i+16, VGPR i-8.
    const bool owner = (lane < 8) || (lane >= 16 && lane < 24);
    const int  trip  = (lane < 8) ? lane : (lane - 8);
    const int  vidx  = trip & 7;
    const float dap2 = diag_pick(accAP, vidx);
    const float dan2 = diag_pick(accAN, vidx);

    float sumv = 0.f, cntv = 0.f;
    if (owner) {
        const int   ti    = t0 + trip;
        const int   ianc  = triplets[3 * ti];
        const float beta  = beta_param[labels_idx[ianc]];
        const float d_ap  = sqrtf(dap2 + EPS_F);
        const float d_an  = sqrtf(dan2 + EPS_F);
        const float pos   = fmaxf(d_ap - beta + MARGIN_F, 0.f);
        const float neg   = fmaxf(beta - d_an + MARGIN_F, 0.f);
        sumv = pos + neg;
        cntv = (pos > 0.f || neg > 0.f) ? 1.f : 0.f;
    }
    sumv = wave_reduce_add(sumv);
    cntv = wave_reduce_add(cntv);

    __shared__ float s_sum[WAVES_PER_BLOCK];
    __shared__ float s_cnt[WAVES_PER_BLOCK];
    if (lane == 0) { s_sum[wave] = sumv; s_cnt[wave] = cntv; }
    __syncthreads();
    if (tid == 0) {
        float bs = 0.f, bc = 0.f;
#pragma unroll
        for (int w = 0; w < WAVES_PER_BLOCK; ++w) { bs += s_sum[w]; bc += s_cnt[w]; }
        partials[2 * blockIdx.x + 0] = bs;
        partials[2 * blockIdx.x + 1] = bc;
    }
}

__global__ __launch_bounds__(256)
void margin_loss_final(const float* __restrict__ partials, int nblocks,
                       float* __restrict__ out)
{
    __shared__ float s_sum[256];
    __shared__ float s_cnt[256];
    const int tid = threadIdx.x;
    float s = 0.f, c = 0.f;
    for (int i = tid; i < nblocks; i += 256) {   // fixed order -> deterministic
        s += partials[2 * i + 0];
        c += partials[2 * i + 1];
    }
    s_sum[tid] = s; s_cnt[tid] = c;
    __syncthreads();
#pragma unroll
    for (int off = 128; off > 0; off >>= 1) {
        if (tid < off) { s_sum[tid] += s_sum[tid + off]; s_cnt[tid] += s_cnt[tid + off]; }
        __syncthreads();
    }
    if (tid == 0) {
        const float total = s_sum[0];
        const float cnt   = s_cnt[0];
        out[0] = (cnt == 0.f) ? total : total / fmaxf(cnt, 1.f);
    }
}

extern "C" void kernel_launch(void* const* d_in, const int* in_sizes, int n_in,
                              void* d_out, int out_size, void* d_ws, size_t ws_size,
                              hipStream_t stream) {
    const float* batch      = (const float*)d_in[0];
    const int*   triplets   = (const int*)  d_in[1];
    const int*   labels_idx = (const int*)  d_in[2];
    const float* beta_param = (const float*)d_in[3];
    float*       out        = (float*)d_out;
    float*       partials   = (float*)d_ws;   // 2 floats per block

    const int T       = in_sizes[1] / 3;            // 262144
    const int nblocks = T / TRIPS_PER_BLOCK;        // 2048 (T divides evenly)

    margin_loss_partial<<<nblocks, BLOCK_THREADS, 0, stream>>>(
        batch, triplets, labels_idx, beta_param, partials);
    margin_loss_final<<<1, 256, 0, stream>>>(partials, nblocks, out);
}